// HairBundleSDE_59914793779563
// MI455X (gfx1250) — compile-verified
//
#include <hip/hip_runtime.h>
#include <hip/hip_bf16.h>
#include <cstdint>

// ---------------------------------------------------------------------------
// HairBundleSDE derivative kernel for MI455X (gfx1250).
// Purely elementwise, bandwidth-bound (~302 MB total traffic -> ~13 us floor
// at 23.3 TB/s). Data path: async global->LDS b128 tile copy (ASYNCcnt),
// conflict-free strided LDS reads, NT gather of force, NT b128 stores.
// ---------------------------------------------------------------------------

typedef float f4 __attribute__((ext_vector_type(4)));
typedef int   i4 __attribute__((ext_vector_type(4)));

// Pointer types matching __builtin_amdgcn_global_load_async_to_lds_b128:
//   (int4 addrspace(1)*, int4 addrspace(3)*, imm offset, imm cpol)
typedef __attribute__((address_space(1))) i4* gas_i4p;
typedef __attribute__((address_space(3))) i4* las_i4p;

#define TPB 256
#define ROWS_PER_BLOCK 1024
#define ROWS_PER_THREAD (ROWS_PER_BLOCK / TPB)          // 4
#define TILE_FLOATS (ROWS_PER_BLOCK * 5)                // 5120 floats = 20 KB
#define VEC_PER_TILE (TILE_FLOATS / 4)                  // 1280 float4
#define VEC_PER_THREAD (VEC_PER_TILE / TPB)             // 5

// exp(U_GS_MAX * DELTA_E) = exp(10.0)
#define E_EXP 22026.465794806718f

#if defined(__has_builtin)
#  if __has_builtin(__builtin_amdgcn_global_load_async_to_lds_b128)
#    define USE_ASYNC_LDS 1
#  else
#    define USE_ASYNC_LDS 0
#  endif
#else
#  define USE_ASYNC_LDS 0
#endif

__device__ __forceinline__ void hb_sde_math(
    float x_hb, float x_a, float p_m, float p_gs, float p_t, float fv,
    float& dx_hb, float& dx_a, float& dp_m, float& dp_gs, float& dp_t)
{
    // Constants from the reference (CHI_HB=1, CHI_A=0.5, X_C=0.3,
    // K_GS_OFFSET=0.5, C_OFFSET=0.9, S_MIN=0.2, S_OFFSET=0.8,
    // S_MAX=0.8, C_MAX=0.2, CA2_M=2, CA2_GS=3, U_GS_MAX=10,
    // TAU_HB=1, TAU_M=0.1, TAU_GS=0.05, TAU_T=0.01)
    const float x_gs = x_hb - 0.5f * x_a + 0.3f;
    const float k_gs = 1.0f - 0.5f * p_gs;
    const float f_gs = k_gs * (x_gs - p_t);

    dx_hb = (-(f_gs + x_hb) + fv);                       // / TAU_HB (=1)

    const float c = 1.0f - 0.9f * p_m;
    const float s = 0.2f + 0.8f * p_m;
    dx_a  = 0.8f * s * (f_gs - x_a) - 0.2f * c;

    dp_m  = (2.0f * p_t * (1.0f - p_m) - p_m) / 0.1f;
    dp_gs = (3.0f * p_t * (1.0f - p_gs) - p_gs) / 0.05f;

    const float arg  = -10.0f * k_gs * (x_gs - 0.5f);
    const float p_t0 = 1.0f / (1.0f + E_EXP * expf(arg));
    dp_t  = (p_t0 - p_t) / 0.01f;
}

__global__ __launch_bounds__(TPB) void hb_sde_tile_kernel(
    const float* __restrict__ x,
    const float* __restrict__ force,
    const int*  __restrict__ t_id_p,
    float* __restrict__ out,
    int f_stride)
{
    __shared__ float tile[TILE_FLOATS];

    const int t = threadIdx.x;
    const long long tile_row0 = (long long)blockIdx.x * ROWS_PER_BLOCK;
    const long long base_f    = tile_row0 * 5;

    const f4* __restrict__ gsrc = (const f4*)(x + base_f);
    f4* __restrict__ lvec = (f4*)tile;

    // ---- stage x tile into LDS with fully-coalesced 128-bit transfers ----
#if USE_ASYNC_LDS
    #pragma unroll
    for (int k = 0; k < VEC_PER_THREAD; ++k) {
        const int idx = t + k * TPB;
        __builtin_amdgcn_global_load_async_to_lds_b128(
            (gas_i4p)(gsrc + idx),
            (las_i4p)(lvec + idx),
            /*offset=*/0, /*cpol=*/0);
    }
    asm volatile("s_wait_asynccnt 0" ::: "memory");
#else
    #pragma unroll
    for (int k = 0; k < VEC_PER_THREAD; ++k) {
        const int idx = t + k * TPB;
        lvec[idx] = gsrc[idx];
    }
#endif
    __syncthreads();

    const int t_id = *t_id_p;   // uniform scalar load

    // ---- compute; each thread owns rows t, t+256, t+512, t+768 ----
    // LDS word index 5*t mod 64 is conflict-free across a wave.
    #pragma unroll
    for (int j = 0; j < ROWS_PER_THREAD; ++j) {
        const int rl   = t + j * TPB;
        const int base = rl * 5;

        const float x_hb = tile[base + 0];
        const float x_a  = tile[base + 1];
        const float p_m  = tile[base + 2];
        const float p_gs = tile[base + 3];
        const float p_t  = tile[base + 4];

        const long long row = tile_row0 + rl;
        const float fv = __builtin_nontemporal_load(
            force + (row * (long long)f_stride + t_id));

        float dx_hb, dx_a, dp_m, dp_gs, dp_t;
        hb_sde_math(x_hb, x_a, p_m, p_gs, p_t, fv,
                    dx_hb, dx_a, dp_m, dp_gs, dp_t);

        // Overwrite our own rows only -> no cross-thread hazard, no barrier
        tile[base + 0] = dx_hb;
        tile[base + 1] = dx_a;
        tile[base + 2] = dp_m;
        tile[base + 3] = dp_gs;
        tile[base + 4] = dp_t;
    }
    __syncthreads();

    // ---- coalesced non-temporal 128-bit stores of the result tile ----
    f4* __restrict__ gdst = (f4*)(out + base_f);
    #pragma unroll
    for (int k = 0; k < VEC_PER_THREAD; ++k) {
        const int idx = t + k * TPB;
        __builtin_nontemporal_store(lvec[idx], gdst + idx);
    }
}

// Generic tail (one row per thread) for row counts not divisible by 1024.
__global__ __launch_bounds__(TPB) void hb_sde_tail_kernel(
    const float* __restrict__ x,
    const float* __restrict__ force,
    const int*  __restrict__ t_id_p,
    float* __restrict__ out,
    int f_stride,
    long long row_start,
    long long n_rows)
{
    const long long row = row_start + blockIdx.x * (long long)TPB + threadIdx.x;
    if (row >= n_rows) return;
    const int t_id = *t_id_p;
    const long long b = row * 5;
    float dx_hb, dx_a, dp_m, dp_gs, dp_t;
    hb_sde_math(x[b + 0], x[b + 1], x[b + 2], x[b + 3], x[b + 4],
                force[row * (long long)f_stride + t_id],
                dx_hb, dx_a, dp_m, dp_gs, dp_t);
    out[b + 0] = dx_hb;
    out[b + 1] = dx_a;
    out[b + 2] = dp_m;
    out[b + 3] = dp_gs;
    out[b + 4] = dp_t;
}

extern "C" void kernel_launch(void* const* d_in, const int* in_sizes, int n_in,
                              void* d_out, int out_size, void* d_ws, size_t ws_size,
                              hipStream_t stream)
{
    const float* x     = (const float*)d_in[0];
    const float* force = (const float*)d_in[1];
    const int*   t_id  = (const int*)d_in[2];
    float*       out   = (float*)d_out;

    const long long n_rows   = (long long)in_sizes[0] / 5;
    const int       f_stride = (n_rows > 0) ? (int)((long long)in_sizes[1] / n_rows) : 8;

    const long long n_tiles = n_rows / ROWS_PER_BLOCK;
    if (n_tiles > 0) {
        hb_sde_tile_kernel<<<(int)n_tiles, TPB, 0, stream>>>(
            x, force, t_id, out, f_stride);
    }
    const long long done = n_tiles * ROWS_PER_BLOCK;
    if (done < n_rows) {
        const long long rem = n_rows - done;
        const int tail_blocks = (int)((rem + TPB - 1) / TPB);
        hb_sde_tail_kernel<<<tail_blocks, TPB, 0, stream>>>(
            x, force, t_id, out, f_stride, done, n_rows);
    }
}